// GATPretrainedWithLinearHead_29446295781607
// MI455X (gfx1250) — compile-verified
//
#include <hip/hip_runtime.h>
#include <hip/hip_bf16.h>
#include <math.h>

#define N_NODES   50000
#define N_EDGES   800000
#define HEADS     4
#define D_HEAD    32
#define FEAT      128            // HEADS*D_HEAD == D_IN
#define OUT_DIM   512
#define NEG_SLOPE 0.2f
#define EPS_      1e-9f

typedef __attribute__((ext_vector_type(2))) float v2f;
typedef __attribute__((ext_vector_type(8))) float v8f;

// ---------------------------------------------------------------- utilities

__global__ void k_zero(float* __restrict__ p, long n) {
    long i = (long)blockIdx.x * blockDim.x + threadIdx.x;
    if (i < n) p[i] = 0.0f;
}

// monotonic float -> u32 key (total order incl. negatives); u32 atomicMax == float max
__device__ __forceinline__ unsigned f2ord(float f) {
    unsigned u = __float_as_uint(f);
    return (u & 0x80000000u) ? ~u : (u | 0x80000000u);
}
__device__ __forceinline__ float ord2f(unsigned k) {
    return (k & 0x80000000u) ? __uint_as_float(k & 0x7fffffffu) : __uint_as_float(~k);
}

// ------------------------------------------------- fp32 WMMA GEMM: C = A*B (+bias)
// A: [M x K] row-major, B: [K x N] row-major, C: [M x N] row-major.
// Block = 128 threads = 4 waves; wave w -> 16x16 tile at (blockIdx.x*4+w)*16, blockIdx.y*16.
// B tile (K x 16) staged in LDS, shared by all 4 waves.
__global__ void k_gemm_wmma(const float* __restrict__ A, const float* __restrict__ B,
                            const float* __restrict__ bias, float* __restrict__ C,
                            int M, int N, int K)
{
    __shared__ float ldsB[128 * 16];             // K <= 128
    const int tid  = threadIdx.x;
    const int wave = tid >> 5;
    const int lane = tid & 31;
    const int n0   = blockIdx.y * 16;

    for (int i = tid; i < K * 16; i += 128) {    // stage B[k][n0..n0+15]
        int k = i >> 4, c = i & 15;
        ldsB[i] = B[(long)k * N + n0 + c];
    }
    __syncthreads();                              // all waves reach barrier before any exit

    const int m0 = (blockIdx.x * 4 + wave) * 16;
    if (m0 >= M) return;                          // wave-uniform: EXEC stays all-1s for WMMA

    const int row   = m0 + (lane & 15);           // A: lane = M row
    const int koff  = (lane >> 4) << 1;           // half-wave selects K pair {0,1} vs {2,3}
    const int col16 = lane & 15;                  // B/C/D: lane = N col

    v8f c = {};
    for (int k0 = 0; k0 < K; k0 += 4) {
        v2f a, b;
        a.x = A[(long)row * K + k0 + koff];
        a.y = A[(long)row * K + k0 + koff + 1];
        b.x = ldsB[(k0 + koff)     * 16 + col16];
        b.y = ldsB[(k0 + koff + 1) * 16 + col16];
        // D = A(16x4 f32) x B(4x16 f32) + C  -> v_wmma_f32_16x16x4_f32
        c = __builtin_amdgcn_wmma_f32_16x16x4_f32(false, a, false, b,
                                                  (short)0, c, false, false);
    }

    const int   col   = n0 + col16;
    const float bv    = bias ? bias[col] : 0.0f;
    const int   rbase = m0 + ((lane >> 4) << 3);  // VGPR i: lanes0-15 -> M=i, lanes16-31 -> M=i+8
#pragma unroll
    for (int i = 0; i < 8; ++i)
        C[(long)(rbase + i) * N + col] = c[i] + bv;
}

// ----------------------------------------------- per-node attention scores
// el[n,h] = <Wh[n,h,:], al[h,:]> ; er likewise
__global__ void k_scores(const float* __restrict__ Wh, const float* __restrict__ al,
                         const float* __restrict__ ar, float* __restrict__ el,
                         float* __restrict__ er, int N)
{
    int idx = blockIdx.x * blockDim.x + threadIdx.x;   // n*HEADS + h
    if (idx >= N * HEADS) return;
    int n = idx >> 2, h = idx & 3;
    const float* w = Wh + (long)n * FEAT + h * D_HEAD;
    const float* a = al + h * D_HEAD;
    const float* b = ar + h * D_HEAD;
    float sl = 0.0f, sr = 0.0f;
#pragma unroll
    for (int d = 0; d < D_HEAD; ++d) {
        sl = fmaf(w[d], a[d], sl);
        sr = fmaf(w[d], b[d], sr);
    }
    el[idx] = sl;
    er[idx] = sr;
}

// --------------------------------- edge pass 1: e = LeakyReLU(el[s]+er[d]); segment max
__global__ void k_edge_pre(const float* __restrict__ el, const float* __restrict__ er,
                           const int* __restrict__ src, const int* __restrict__ dst,
                           float* __restrict__ ebuf, unsigned* __restrict__ menc, int E)
{
    int e = blockIdx.x * blockDim.x + threadIdx.x;
    if (e >= E) return;
    int s = src[e], d = dst[e];
#pragma unroll
    for (int h = 0; h < HEADS; ++h) {
        float v = el[s * HEADS + h] + er[d * HEADS + h];
        v = (v > 0.0f) ? v : NEG_SLOPE * v;
        ebuf[(long)e * HEADS + h] = v;
        atomicMax(&menc[d * HEADS + h], f2ord(v));
    }
}

// --------------------------------- edge pass 2: ex = exp(e - m[dst]); segment sum
__global__ void k_edge_softmax(const int* __restrict__ dst, const unsigned* __restrict__ menc,
                               float* __restrict__ ebuf, float* __restrict__ denom, int E)
{
    int e = blockIdx.x * blockDim.x + threadIdx.x;
    if (e >= E) return;
    int d = dst[e];
#pragma unroll
    for (int h = 0; h < HEADS; ++h) {
        float m  = ord2f(menc[d * HEADS + h]);     // only read for dst with >=1 edge
        float ex = expf(ebuf[(long)e * HEADS + h] - m);
        ebuf[(long)e * HEADS + h] = ex;
        atomicAdd(&denom[d * HEADS + h], ex);
    }
}

// --------------------------------- edge pass 3: acc[dst] += alpha * Wh[src]
// one 128-thread block per edge; thread = feature index
__global__ void k_edge_scatter(const float* __restrict__ Wh, const float* __restrict__ ex,
                               const float* __restrict__ denom, const int* __restrict__ src,
                               const int* __restrict__ dst, float* __restrict__ acc, int E)
{
    int e = blockIdx.x;
    int f = threadIdx.x;                 // 0..127
    int h = f >> 5;
    int s = src[e], d = dst[e];
    float alpha = ex[(long)e * HEADS + h] / (denom[d * HEADS + h] + EPS_);
    atomicAdd(&acc[(long)d * FEAT + f], alpha * Wh[(long)s * FEAT + f]);
}

// --------------------------------- layer-1 finalize: z = ELU(acc + b)   (in place)
__global__ void k_elu(float* __restrict__ z, const float* __restrict__ b, long n)
{
    long i = (long)blockIdx.x * blockDim.x + threadIdx.x;
    if (i >= n) return;
    float x = z[i] + b[i & (FEAT - 1)];
    z[i] = (x > 0.0f) ? x : expm1f(x);
}

// --------------------------------- layer-2 finalize: zmean[n,d] = mean_h(acc[n,h,d]+b[h,d])
__global__ void k_mean(const float* __restrict__ acc, const float* __restrict__ b,
                       float* __restrict__ zmean, int N)
{
    int i = blockIdx.x * blockDim.x + threadIdx.x;   // n*D_HEAD + d
    if (i >= N * D_HEAD) return;
    int n = i >> 5, d = i & 31;
    float s = 0.0f;
#pragma unroll
    for (int h = 0; h < HEADS; ++h)
        s += acc[(long)n * FEAT + h * D_HEAD + d] + b[h * D_HEAD + d];
    zmean[i] = s * (1.0f / HEADS);
}

// ============================================================== launcher

extern "C" void kernel_launch(void* const* d_in, const int* in_sizes, int n_in,
                              void* d_out, int out_size, void* d_ws, size_t ws_size,
                              hipStream_t stream)
{
    const float* h   = (const float*)d_in[0];
    const int*   src = (const int*)  d_in[1];
    const int*   dst = (const int*)  d_in[2];
    const float* W1  = (const float*)d_in[3];
    const float* al1 = (const float*)d_in[4];
    const float* ar1 = (const float*)d_in[5];
    const float* b1  = (const float*)d_in[6];
    const float* W2  = (const float*)d_in[7];
    const float* al2 = (const float*)d_in[8];
    const float* ar2 = (const float*)d_in[9];
    const float* b2  = (const float*)d_in[10];
    const float* Wp  = (const float*)d_in[11];
    const float* bp  = (const float*)d_in[12];
    float*       out = (float*)d_out;

    const int N = N_NODES, E = N_EDGES;

    // workspace carve-up (~74 MB)
    float*    Wh    = (float*)d_ws;                       // N*FEAT
    float*    Z     = Wh + (size_t)N * FEAT;              // N*FEAT  (acc / z)
    float*    el    = Z  + (size_t)N * FEAT;              // N*HEADS
    float*    er    = el + (size_t)N * HEADS;             // N*HEADS
    unsigned* menc  = (unsigned*)(er + (size_t)N * HEADS);// N*HEADS
    float*    denom = (float*)(menc + (size_t)N * HEADS); // N*HEADS
    float*    ebuf  = denom + (size_t)N * HEADS;          // E*HEADS
    float*    zmean = ebuf + (size_t)E * HEADS;           // N*D_HEAD

    const int mtiles = (N + 15) / 16;                     // 3125
    const dim3 gemmBlk(128);
    const dim3 gemmGrd128((mtiles + 3) / 4, FEAT / 16);   // (782, 8)  K=128,N=128
    const dim3 gemmGrdOut((mtiles + 3) / 4, OUT_DIM / 16);// (782, 32) K=32, N=512

    const int zThreads = 256;
    auto zgrid = [](long n) { return (unsigned)((n + 255) / 256); };

    // ---------------- layer 1 ----------------
    k_gemm_wmma<<<gemmGrd128, gemmBlk, 0, stream>>>(h, W1, nullptr, Wh, N, FEAT, FEAT);
    k_scores<<<(N * HEADS + 255) / 256, 256, 0, stream>>>(Wh, al1, ar1, el, er, N);
    k_zero<<<zgrid((long)N * HEADS), zThreads, 0, stream>>>((float*)menc, (long)N * HEADS);
    k_zero<<<zgrid((long)N * HEADS), zThreads, 0, stream>>>(denom, (long)N * HEADS);
    k_zero<<<zgrid((long)N * FEAT), zThreads, 0, stream>>>(Z, (long)N * FEAT);
    k_edge_pre<<<(E + 255) / 256, 256, 0, stream>>>(el, er, src, dst, ebuf, menc, E);
    k_edge_softmax<<<(E + 255) / 256, 256, 0, stream>>>(dst, menc, ebuf, denom, E);
    k_edge_scatter<<<E, FEAT, 0, stream>>>(Wh, ebuf, denom, src, dst, Z, E);
    k_elu<<<zgrid((long)N * FEAT), zThreads, 0, stream>>>(Z, b1, (long)N * FEAT);

    // ---------------- layer 2 ----------------
    k_gemm_wmma<<<gemmGrd128, gemmBlk, 0, stream>>>(Z, W2, nullptr, Wh, N, FEAT, FEAT);
    k_scores<<<(N * HEADS + 255) / 256, 256, 0, stream>>>(Wh, al2, ar2, el, er, N);
    k_zero<<<zgrid((long)N * HEADS), zThreads, 0, stream>>>((float*)menc, (long)N * HEADS);
    k_zero<<<zgrid((long)N * HEADS), zThreads, 0, stream>>>(denom, (long)N * HEADS);
    k_zero<<<zgrid((long)N * FEAT), zThreads, 0, stream>>>(Z, (long)N * FEAT);
    k_edge_pre<<<(E + 255) / 256, 256, 0, stream>>>(el, er, src, dst, ebuf, menc, E);
    k_edge_softmax<<<(E + 255) / 256, 256, 0, stream>>>(dst, menc, ebuf, denom, E);
    k_edge_scatter<<<E, FEAT, 0, stream>>>(Wh, ebuf, denom, src, dst, Z, E);
    k_mean<<<(N * D_HEAD + 255) / 256, 256, 0, stream>>>(Z, b2, zmean, N);

    // ---------------- projection head ----------------
    k_gemm_wmma<<<gemmGrdOut, gemmBlk, 0, stream>>>(zmean, Wp, bp, out, N, OUT_DIM, D_HEAD);
}